// ChannelProp_85341000172341
// MI455X (gfx1250) — compile-verified
//
#include <hip/hip_runtime.h>
#include <stdint.h>

#define D_FEAT 96
#define TPB    256          // 8 waves of 32 (wave32 on gfx1250)
#define EPB    256          // edges staged per block (one per thread)

// ---- CDNA5 async global->LDS path (guarded: falls back cleanly) -----------
#if defined(__AMDGCN__) && __has_builtin(__builtin_amdgcn_global_load_async_to_lds_b32)
#define USE_ASYNC_LDS 1
#else
#define USE_ASYNC_LDS 0
#endif

typedef __attribute__((address_space(1))) int glb_i32;   // global int*
typedef __attribute__((address_space(3))) int lds_i32;   // LDS int*

__device__ __forceinline__ void atomic_add_f32(float* p, float v) {
    // relaxed, device(agent) scope -> native global_atomic_add_f32 (no CAS loop)
    __hip_atomic_fetch_add(p, v, __ATOMIC_RELAXED, __HIP_MEMORY_SCOPE_AGENT);
}
__device__ __forceinline__ void atomic_add_i32(int* p, int v) {
    __hip_atomic_fetch_add(p, v, __ATOMIC_RELAXED, __HIP_MEMORY_SCOPE_AGENT);
}

// ---------------------------------------------------------------------------
// MAIN kernel (placed first so the disasm snippet shows it):
// gather/scale/scatter-add. One wave per edge iteration; lane L handles
// features L, L+32, L+64 (fully coalesced dword accesses).
__global__ __launch_bounds__(TPB) void cp_scatter_kernel(
    const float* __restrict__ x, const int* __restrict__ row,
    const int* __restrict__ col, const float* __restrict__ dis,
    float* __restrict__ out, int n_edges) {
    __shared__ int s_row[EPB];
    __shared__ int s_col[EPB];
    const int t    = threadIdx.x;
    const int base = blockIdx.x * EPB;

    // ---- stage this block's edge indices into LDS (ASYNCcnt path) ---------
    {
        int e     = base + t;
        int esafe = (e < n_edges) ? e : 0;
#if USE_ASYNC_LDS
        __builtin_amdgcn_global_load_async_to_lds_b32(
            (glb_i32*)(int*)(row + esafe), (lds_i32*)&s_row[t], 0, 0);
        __builtin_amdgcn_global_load_async_to_lds_b32(
            (glb_i32*)(int*)(col + esafe), (lds_i32*)&s_col[t], 0, 0);
#if __has_builtin(__builtin_amdgcn_s_wait_asynccnt)
        __builtin_amdgcn_s_wait_asynccnt(0);
#else
        asm volatile("s_wait_asynccnt 0" ::: "memory");
#endif
#else
        s_row[t] = row[esafe];
        s_col[t] = col[esafe];
#endif
    }
    __syncthreads();

    const int wave = t >> 5;            // wave32
    const int lane = t & 31;
    const int edges_per_wave = EPB / (TPB / 32);   // 32

    for (int i = 0; i < edges_per_wave; ++i) {
        int el = wave * edges_per_wave + i;
        int e  = base + el;
        if (e >= n_edges) break;                   // wave-uniform

        // r/c are wave-uniform (all lanes read the same LDS word):
        // readfirstlane makes that provable -> SMEM loads for dis[],
        // SGPR base addresses for the x / out rows.
        int r = __builtin_amdgcn_readfirstlane(s_row[el]);
        int c = __builtin_amdgcn_readfirstlane(s_col[el]);
        float norm = dis[r] * dis[c];

        // prefetch next edge's feature row (lane*3 stride covers all 384B)
        if (i + 1 < edges_per_wave) {
            int cn = __builtin_amdgcn_readfirstlane(s_col[el + 1]);
            __builtin_prefetch(x + (size_t)cn * D_FEAT + lane * 3, 0, 0);
        }

        const float* xr   = x   + (size_t)c * D_FEAT;
        float*       orow = out + (size_t)r * D_FEAT;
        float a0 = xr[lane];
        float a1 = xr[lane + 32];
        float a2 = xr[lane + 64];
        atomic_add_f32(&orow[lane],      a0 * norm);
        atomic_add_f32(&orow[lane + 32], a1 * norm);
        atomic_add_f32(&orow[lane + 64], a2 * norm);
    }
}

// ---------------------------------------------------------------------------
// Kernel: zero output and degree counters
__global__ void cp_zero_kernel(float* __restrict__ out, int* __restrict__ cnt,
                               int n_out, int n_cnt) {
    int i = blockIdx.x * blockDim.x + threadIdx.x;
    int stride = gridDim.x * blockDim.x;
    for (int j = i; j < n_out; j += stride) out[j] = 0.0f;
    for (int j = i; j < n_cnt; j += stride) cnt[j] = 0;
}

// Kernel: degree bincount over destination (row) indices
__global__ void cp_degree_kernel(const int* __restrict__ row,
                                 int* __restrict__ cnt, int n_edges) {
    int e = blockIdx.x * blockDim.x + threadIdx.x;
    if (e < n_edges) atomic_add_i32(&cnt[row[e]], 1);
}

// Kernel: deg_inv_sqrt, in place in the workspace (int count -> float)
__global__ void cp_dis_kernel(void* ws, int n_nodes) {
    int i = blockIdx.x * blockDim.x + threadIdx.x;
    if (i < n_nodes) {
        int c = ((int*)ws)[i];
        ((float*)ws)[i] = (c > 0) ? rsqrtf((float)c) : 0.0f;
    }
}

// ---------------------------------------------------------------------------
extern "C" void kernel_launch(void* const* d_in, const int* in_sizes, int n_in,
                              void* d_out, int out_size, void* d_ws, size_t ws_size,
                              hipStream_t stream) {
    const float* x  = (const float*)d_in[0];
    const int*   ei = (const int*)d_in[1];
    const int n_nodes = in_sizes[0] / D_FEAT;
    const int n_edges = in_sizes[1] / 2;
    const int* row = ei;                 // edge_index[0]
    const int* col = ei + n_edges;       // edge_index[1]
    float* out = (float*)d_out;
    int*   cnt = (int*)d_ws;             // n_nodes ints, then reused as float dis
    float* dis = (float*)d_ws;

    const int n_out = n_nodes * D_FEAT;

    cp_zero_kernel<<<2048, TPB, 0, stream>>>(out, cnt, n_out, n_nodes);
    cp_degree_kernel<<<(n_edges + TPB - 1) / TPB, TPB, 0, stream>>>(row, cnt, n_edges);
    cp_dis_kernel<<<(n_nodes + TPB - 1) / TPB, TPB, 0, stream>>>(d_ws, n_nodes);

    const int nblocks = (n_edges + EPB - 1) / EPB;
    cp_scatter_kernel<<<nblocks, TPB, 0, stream>>>(x, row, col, dis, out, n_edges);
}